// VectorQuantizer_org_VQGAN_16329465659955
// MI455X (gfx1250) — compile-verified
//
#include <hip/hip_runtime.h>
#include <hip/hip_bf16.h>

// Problem constants (from reference): z (32,256,32,32) f32, emb (1024,256) f32
#define BATCH   32
#define CHAN    256
#define HW      1024            // 32*32
#define NROWS   32768           // BATCH*HW
#define NE      1024            // codebook entries
#define EDIM    256
#define CHW     262144          // CHAN*HW
#define ZQ_ELEMS 8388608        // 32*256*32*32
#define LOSS_OFF 8388608
#define IDX_OFF  8388609

#define MTILE   64              // rows per block (4 WMMA M-tiles) -> 4x B reuse
#define NWAVES  8               // 256 threads
#define ASTRIDE 260             // padded LDS row stride (floats), 8B-aligned

typedef __attribute__((ext_vector_type(2))) float v2f;
typedef __attribute__((ext_vector_type(8))) float v8f;
typedef unsigned long long u64;

// Sortable key: monotone float->uint transform in high 32 bits, code in low
// 32 bits. u64-min == (min dist, then min code) — jnp.argmin tie-break.
__device__ __forceinline__ u64 vq_key(float d, int code) {
    unsigned u = __float_as_uint(d);
    u ^= (unsigned)((int)u >> 31) | 0x80000000u;
    return ((u64)u << 32) | (unsigned)code;
}
__device__ __forceinline__ u64 u64min(u64 a, u64 b) { return a < b ? a : b; }

// ---------------- Kernel 0: codebook norms + zero loss slot ----------------
__global__ void vq_enorm_kernel(const float* __restrict__ emb,
                                float* __restrict__ enorm,
                                float* __restrict__ loss_ptr) {
    int row  = blockIdx.x * 8 + (threadIdx.x >> 5);   // one wave32 per codebook row
    int lane = threadIdx.x & 31;
    const float* p = emb + (size_t)row * EDIM;
    float s = 0.0f;
    for (int c = lane; c < EDIM; c += 32) { float v = p[c]; s += v * v; }
    for (int off = 16; off; off >>= 1) s += __shfl_xor(s, off, 32);
    if (lane == 0) enorm[row] = s;
    if (blockIdx.x == 0 && threadIdx.x == 0) *loss_ptr = 0.0f;
}

// ---------------- Main fused VQ kernel ----------------
// grid: 512 blocks (one per 64-row supertile), block: 256 threads = 8 waves.
// Each B (codebook) tile load from L2 feeds 4 WMMAs (4 M-tiles).
__global__ void __launch_bounds__(256)
vq_main_kernel(const float* __restrict__ z,
               const float* __restrict__ emb,
               const float* __restrict__ enorm,
               float* __restrict__ out) {
    __shared__ float A_lds[MTILE * ASTRIDE];   // zf supertile, 64 rows x 256 chans
    __shared__ float znorm_lds[MTILE];
    __shared__ u64   red_key[NWAVES][MTILE];
    __shared__ int   idx_lds[MTILE];
    __shared__ float loss_red[NWAVES * 32];

    const int n0  = blockIdx.x * MTILE;     // first global row of this supertile
    const int b   = n0 >> 10;               // batch index (HW==1024, 64 | 1024)
    const int hw0 = n0 & (HW - 1);          // hw offset (no batch crossing)

    // ---- Stage zf supertile into LDS. Rows are consecutive hw -> for a fixed
    // c the 64 row-values are 256B contiguous in z. ----
    const float* zb = z + (size_t)b * CHW + hw0;
    for (int i = threadIdx.x; i < MTILE * CHAN; i += NWAVES * 32) {
        int c = i >> 6;                     // 0..255
        int m = i & (MTILE - 1);            // 0..63
        A_lds[m * ASTRIDE + c] = zb[(size_t)c * HW + m];
    }
    __syncthreads();

    // ---- Per-row ||z||^2 (64 rows, one thread each) ----
    if (threadIdx.x < MTILE) {
        float s = 0.0f;
        const float* ar = &A_lds[threadIdx.x * ASTRIDE];
        for (int c = 0; c < CHAN; ++c) { float v = ar[c]; s += v * v; }
        znorm_lds[threadIdx.x] = s;
    }
    __syncthreads();

    const int lane  = threadIdx.x & 31;
    const int wv    = threadIdx.x >> 5;
    const int mn    = lane & 15;            // A: M row (within tile); B: N col
    const int khalf = (lane >> 4) << 1;     // lanes 16-31 carry K+2,K+3
    const int mhi   = (lane >> 4) << 3;     // C/D: +8 rows for upper lane half

    u64 bestk[4][8];
#pragma unroll
    for (int t = 0; t < 4; ++t)
#pragma unroll
        for (int r = 0; r < 8; ++r) bestk[t][r] = ~0ull;

    // ---- Sweep codebook: 64 tiles of 16 codes, 8 tiles per wave ----
    for (int kt = wv; kt < NE / 16; kt += NWAVES) {
        const int kb = kt * 16;
        v8f acc0 = {}, acc1 = {}, acc2 = {}, acc3 = {};
        const float* embp = emb + (size_t)(kb + mn) * EDIM + khalf;
        const float* ap   = &A_lds[mn * ASTRIDE + khalf];
#pragma unroll 4
        for (int kk = 0; kk < EDIM; kk += 4) {
            v2f bb = *(const v2f*)(embp + kk);                     // 1 L2 load ...
            v2f a0 = *(const v2f*)(ap + kk);                       // ds_load_b64
            v2f a1 = *(const v2f*)(ap + 16 * ASTRIDE + kk);
            v2f a2 = *(const v2f*)(ap + 32 * ASTRIDE + kk);
            v2f a3 = *(const v2f*)(ap + 48 * ASTRIDE + kk);
            // ... feeds 4 independent WMMAs (no D->A/B hazard chain)
            acc0 = __builtin_amdgcn_wmma_f32_16x16x4_f32(false, a0, false, bb, (short)0, acc0, false, false);
            acc1 = __builtin_amdgcn_wmma_f32_16x16x4_f32(false, a1, false, bb, (short)0, acc1, false, false);
            acc2 = __builtin_amdgcn_wmma_f32_16x16x4_f32(false, a2, false, bb, (short)0, acc2, false, false);
            acc3 = __builtin_amdgcn_wmma_f32_16x16x4_f32(false, a3, false, bb, (short)0, acc3, false, false);
        }
        const float en   = enorm[kb + mn];          // code = kb + N(lane)
        const int   code = kb + mn;
#pragma unroll
        for (int r = 0; r < 8; ++r) {               // branchless u64-min argmin
            float d0 = znorm_lds[ 0 + r + mhi] + en - 2.0f * acc0[r];
            float d1 = znorm_lds[16 + r + mhi] + en - 2.0f * acc1[r];
            float d2 = znorm_lds[32 + r + mhi] + en - 2.0f * acc2[r];
            float d3 = znorm_lds[48 + r + mhi] + en - 2.0f * acc3[r];
            bestk[0][r] = u64min(bestk[0][r], vq_key(d0, code));
            bestk[1][r] = u64min(bestk[1][r], vq_key(d1, code));
            bestk[2][r] = u64min(bestk[2][r], vq_key(d2, code));
            bestk[3][r] = u64min(bestk[3][r], vq_key(d3, code));
        }
    }

    // ---- Cross-lane argmin within each 16-lane group (N dimension) ----
#pragma unroll
    for (int t = 0; t < 4; ++t) {
#pragma unroll
        for (int r = 0; r < 8; ++r) {
            u64 k = bestk[t][r];
            for (int off = 8; off; off >>= 1)
                k = u64min(k, (u64)__shfl_xor((long long)k, off, 16));
            if ((lane & 15) == 0)
                red_key[wv][t * 16 + r + mhi] = k;
        }
    }
    __syncthreads();

    // ---- Combine the 8 waves; emit indices ----
    if (threadIdx.x < MTILE) {
        int m = threadIdx.x;
        u64 k = red_key[0][m];
#pragma unroll
        for (int w = 1; w < NWAVES; ++w) k = u64min(k, red_key[w][m]);
        int ix = (int)(unsigned)(k & 0xffffffffull);
        idx_lds[m] = ix;
        out[IDX_OFF + n0 + m] = (float)ix;   // z_indices (f32 out buffer)
    }
    __syncthreads();

    // ---- z_q gather + NCHW scatter + loss partial ----
    // STE forward value equals z_q; loss = 1.25 * mean((z_q - z)^2).
    float lsum = 0.0f;
    float* outb = out + (size_t)b * CHW + hw0;
    for (int i = threadIdx.x; i < MTILE * CHAN; i += NWAVES * 32) {
        int c = i >> 6;
        int m = i & (MTILE - 1);
        float q  = emb[(size_t)idx_lds[m] * EDIM + c];
        float zv = A_lds[m * ASTRIDE + c];
        float d  = q - zv;
        lsum += d * d;
        outb[(size_t)c * HW + m] = q;        // 256B contiguous segments
    }
    loss_red[threadIdx.x] = lsum;
    __syncthreads();
    for (int s = 128; s; s >>= 1) {
        if (threadIdx.x < s) loss_red[threadIdx.x] += loss_red[threadIdx.x + s];
        __syncthreads();
    }
    if (threadIdx.x == 0)
        atomicAdd(out + LOSS_OFF, loss_red[0] * (1.25f / (float)ZQ_ELEMS));
}

extern "C" void kernel_launch(void* const* d_in, const int* in_sizes, int n_in,
                              void* d_out, int out_size, void* d_ws, size_t ws_size,
                              hipStream_t stream) {
    const float* z   = (const float*)d_in[0];   // (32,256,32,32)
    const float* emb = (const float*)d_in[1];   // (1024,256)
    float* out   = (float*)d_out;               // [z_q | loss | indices]
    float* enorm = (float*)d_ws;                // 1024 floats of scratch

    vq_enorm_kernel<<<NE / 8, 256, 0, stream>>>(emb, enorm, out + LOSS_OFF);
    vq_main_kernel<<<NROWS / MTILE, 256, 0, stream>>>(z, emb, enorm, out);
}